// MeanPoolDownsample_69827578298481
// MI455X (gfx1250) — compile-verified
//
#include <hip/hip_runtime.h>

#ifndef D_FEAT
#define D_FEAT 128
#endif

// Native clang vector type: required by __builtin_nontemporal_load/store,
// lowers to b128 global memory ops.
typedef float v4f __attribute__((ext_vector_type(4)));

// ---------------------------------------------------------------------------
// Kernel 1: zero-init sums (d_out) and counts (d_ws).
// Grid-stride, b128 stores for the big buffer.
// ---------------------------------------------------------------------------
__global__ void mp_zero_kernel(v4f* __restrict__ sums4, long n_sums4,
                               float* __restrict__ counts, int n_counts) {
    const long stride = (long)gridDim.x * blockDim.x;
    long i = (long)blockIdx.x * blockDim.x + threadIdx.x;
    const v4f z = {0.f, 0.f, 0.f, 0.f};
    for (long j = i; j < n_sums4; j += stride) {
        sums4[j] = z;
    }
    for (long j = i; j < (long)n_counts; j += stride) {
        counts[j] = 0.f;
    }
}

// ---------------------------------------------------------------------------
// Kernel 2: atomic scatter-add.
// One wave32 per fine row: lane l handles floats [4l, 4l+4).
//  - input row load: one fully-coalesced 512B transaction per wave,
//    NON-TEMPORAL so the 1 GB one-shot stream doesn't evict the ~258 MB
//    scatter working set from the 192 MB L2 (atomics want to resolve in L2).
//  - 4x GLOBAL_ATOMIC_ADD_F32 (no return -> STOREcnt path) per lane.
//  - lane 0 adds 1.0 to counts[seg].
// ---------------------------------------------------------------------------
__global__ void mp_scatter_kernel(const float* __restrict__ x,
                                  const int* __restrict__ seg_ids,
                                  float* __restrict__ sums,
                                  float* __restrict__ counts,
                                  int n_fine) {
    const int gtid = blockIdx.x * blockDim.x + threadIdx.x;
    const int row  = gtid >> 5;   // wave32: one row per wave
    const int lane = gtid & 31;
    if (row >= n_fine) return;

    const int s = seg_ids[row];   // wave-uniform -> scalar path

    // Streaming, use-once input: non-temporal b128 load.
    const v4f* row_ptr = (const v4f*)(x + (long)row * D_FEAT) + lane;
    const v4f v = __builtin_nontemporal_load(row_ptr);

    float* dst = sums + (long)s * D_FEAT + (lane << 2);
    // Relaxed agent-scope fadd with discarded return -> hardware
    // global_atomic_add_f32 (non-returning).
    (void)__hip_atomic_fetch_add(dst + 0, v.x, __ATOMIC_RELAXED,
                                 __HIP_MEMORY_SCOPE_AGENT);
    (void)__hip_atomic_fetch_add(dst + 1, v.y, __ATOMIC_RELAXED,
                                 __HIP_MEMORY_SCOPE_AGENT);
    (void)__hip_atomic_fetch_add(dst + 2, v.z, __ATOMIC_RELAXED,
                                 __HIP_MEMORY_SCOPE_AGENT);
    (void)__hip_atomic_fetch_add(dst + 3, v.w, __ATOMIC_RELAXED,
                                 __HIP_MEMORY_SCOPE_AGENT);

    if (lane == 0) {
        (void)__hip_atomic_fetch_add(counts + s, 1.0f, __ATOMIC_RELAXED,
                                     __HIP_MEMORY_SCOPE_AGENT);
    }
}

// ---------------------------------------------------------------------------
// Kernel 3: finalize in place: out = count>0 ? sum/count : 0.
// One wave32 per segment, v4f per lane (b128 load + NT b128 store).
// ---------------------------------------------------------------------------
__global__ void mp_finalize_kernel(float* __restrict__ sums,
                                   const float* __restrict__ counts,
                                   int n_coarse) {
    const int gtid = blockIdx.x * blockDim.x + threadIdx.x;
    const int seg  = gtid >> 5;
    const int lane = gtid & 31;
    if (seg >= n_coarse) return;

    const float c = counts[seg];            // wave-uniform
    const float inv = (c > 0.f) ? (1.0f / c) : 0.0f;

    v4f* p = (v4f*)(sums + (long)seg * D_FEAT) + lane;
    v4f v = *p;
    v.x *= inv; v.y *= inv; v.z *= inv; v.w *= inv;
    // Result is consumed by the host, not re-read by us: non-temporal store.
    __builtin_nontemporal_store(v, p);
}

// ---------------------------------------------------------------------------
// Launch wrapper
//   d_in[0] : fine_scale_h_d            float32 [N_FINE, 128]
//   d_in[1] : prolongation_map          int32   [N_FINE]
//   d_in[2] : target_num_segments       int32   [1] (scalar; value == out rows)
//   d_out   : float32 [N_COARSE, 128]
//   d_ws    : >= N_COARSE * 4 bytes (counts)
// ---------------------------------------------------------------------------
extern "C" void kernel_launch(void* const* d_in, const int* in_sizes, int n_in,
                              void* d_out, int out_size, void* d_ws, size_t ws_size,
                              hipStream_t stream) {
    const float* x       = (const float*)d_in[0];
    const int*   seg_ids = (const int*)d_in[1];
    float*       sums    = (float*)d_out;
    float*       counts  = (float*)d_ws;

    const int n_fine   = in_sizes[0] / D_FEAT;   // 2,000,000
    const int n_coarse = out_size / D_FEAT;      // 500,000

    // --- 1) zero sums + counts -------------------------------------------
    {
        const long n_sums4 = (long)n_coarse * (D_FEAT / 4);
        const int threads = 256;
        const int blocks  = 4096;   // grid-stride
        mp_zero_kernel<<<blocks, threads, 0, stream>>>(
            (v4f*)sums, n_sums4, counts, n_coarse);
    }

    // --- 2) atomic scatter-add: one wave per row -------------------------
    {
        const int threads = 256;                       // 8 waves/block
        const long total  = (long)n_fine * 32;         // 32 lanes per row
        const int blocks  = (int)((total + threads - 1) / threads);
        mp_scatter_kernel<<<blocks, threads, 0, stream>>>(
            x, seg_ids, sums, counts, n_fine);
    }

    // --- 3) finalize: one wave per segment -------------------------------
    {
        const int threads = 256;
        const long total  = (long)n_coarse * 32;
        const int blocks  = (int)((total + threads - 1) / threads);
        mp_finalize_kernel<<<blocks, threads, 0, stream>>>(
            sums, counts, n_coarse);
    }
}